// ResonantComplexProjection_75892072121060
// MI455X (gfx1250) — compile-verified
//
#include <hip/hip_runtime.h>
#include <hip/hip_bf16.h>
#include <math.h>

typedef float v2f __attribute__((ext_vector_type(2)));
typedef float v8f __attribute__((ext_vector_type(8)));

constexpr int BATCH = 256;
constexpr int IN_F  = 512;
constexpr int OUT_F = 512;
constexpr int NNEUR = 1024;
constexpr int LUT   = 4096;

// f32(2*pi/4096) and f32(4096/(2*pi)) -- match jnp's f64-scalar -> f32 promotion
#define K_ANG   1.5339807878856412e-3f
#define K_SCALE 651.8986469044033f

// ---------------------------------------------------------------------------
// Kernel 0: build the sin/cos LUT once into workspace.
// lut[0..4095] = sin, lut[4096..8191] = cos.
// ---------------------------------------------------------------------------
__global__ __launch_bounds__(256) void build_lut_kernel(float* __restrict__ lut)
{
    const int t = blockIdx.x * 256 + threadIdx.x;   // 0..4095
    const float ang = (float)t * K_ANG;
    lut[t]       = sinf(ang);
    lut[t + LUT] = cosf(ang);
}

// ---------------------------------------------------------------------------
// Kernel 1: cos_sum / sin_sum over the in_features axis.
// One block = one batch row b x 64 neurons (8 waves x 8 neurons each).
// LUT staged from global into LDS with float4 copies (amortized over 8
// neurons per wave); x-row staged too. Lanes stride i (coalesced W/B reads),
// wave32 shfl_xor reduction per neuron.
// ---------------------------------------------------------------------------
__global__ __launch_bounds__(256) void resonant_sums_kernel(
    const float* __restrict__ x, const float* __restrict__ W,
    const float* __restrict__ B, const float* __restrict__ lut,
    float* __restrict__ Scos, float* __restrict__ Ssin)
{
    __shared__ float lutSC[2 * LUT];   // [0..4095]=sin, [4096..8191]=cos
    __shared__ float xrow[IN_F];

    const int tid = threadIdx.x;
    const int b   = blockIdx.y;

    // Stage 32KB LUT: 2048 float4 / 256 threads = 8 b128 copies per thread.
    {
        const float4* __restrict__ src = (const float4*)lut;
        float4* dst = (float4*)lutSC;
        for (int t = tid; t < (2 * LUT) / 4; t += 256) dst[t] = src[t];
    }
    // Stage x row (512 floats = 128 float4).
    if (tid < IN_F / 4)
        ((float4*)xrow)[tid] = ((const float4*)(x + (size_t)b * IN_F))[tid];
    __syncthreads();

    const int lane  = tid & 31;
    const int wave  = tid >> 5;
    const int nBase = blockIdx.x * 64 + wave * 8;

    for (int j = 0; j < 8; ++j) {
        const int n = nBase + j;
        const float* __restrict__ Wn = W + (size_t)n * IN_F;
        const float* __restrict__ Bn = B + (size_t)n * IN_F;

        float accS = 0.0f, accC = 0.0f;
        for (int i0 = 0; i0 < IN_F; i0 += 32) {
            const int i    = i0 + lane;
            const float wl = 1.0f + fabsf(Wn[i]);            // wavelength
            const float th = xrow[i] / wl + Bn[i];           // IEEE div, as ref
            const int idx  = ((int)floorf(th * K_SCALE)) & (LUT - 1); // mod 4096
            accS += lutSC[idx];
            accC += lutSC[LUT + idx];
        }

        // wave32 butterfly reduction
        for (int off = 16; off >= 1; off >>= 1) {
            accS += __shfl_xor(accS, off, 32);
            accC += __shfl_xor(accC, off, 32);
        }
        if (lane == 0) {
            Scos[(size_t)b * NNEUR + n] = accC;
            Ssin[(size_t)b * NNEUR + n] = accS;
        }
    }
}

// ---------------------------------------------------------------------------
// Kernel 2: real = cos_sum @ Wr^T, imag = sin_sum @ Wi^T via
// V_WMMA_F32_16X16X4_F32. One wave per 16x16 output tile; both GEMMs share
// the tile loop (two wmma issues per K-step).
//
// ISA VGPR layouts (CDNA5 ISA):
//   A 16x4 : lane l<16 -> row l, K=(k0,k0+1);  lane>=16 -> row l-16, K=(k0+2,k0+3)
//   B 4x16 : lane l<16 -> col l, K=(k0,k0+1);  lane>=16 -> col l-16, K=(k0+2,k0+3)
//   C 16x16: vgpr r -> row r + 8*(lane>=16), col = lane&15
// Both A and B fragments are contiguous float2 per lane.
// ---------------------------------------------------------------------------
__global__ __launch_bounds__(128) void resonant_gemm_kernel(
    const float* __restrict__ Scos, const float* __restrict__ Ssin,
    const float* __restrict__ Wr,   const float* __restrict__ Wi,
    float* __restrict__ outReal,    float* __restrict__ outImag)
{
    const int lane  = threadIdx.x & 31;
    const int wave  = threadIdx.x >> 5;
    const int nTile = blockIdx.x * 4 + wave;   // output-column tile
    const int mTile = blockIdx.y;              // output-row (batch) tile
    const int half  = lane >> 4;               // 0: K lo pair, 1: K hi pair
    const int l15   = lane & 15;

    const int m = mTile * 16 + l15;            // A row (batch)
    const int o = nTile * 16 + l15;            // B col (out feature)

    const float* __restrict__ aC = Scos + (size_t)m * NNEUR;
    const float* __restrict__ aS = Ssin + (size_t)m * NNEUR;
    const float* __restrict__ bR = Wr   + (size_t)o * NNEUR;  // Wr[o][k] contiguous
    const float* __restrict__ bI = Wi   + (size_t)o * NNEUR;

    v8f cr = {};
    v8f ci = {};

    for (int k0 = 0; k0 < NNEUR; k0 += 4) {
        const int kk = k0 + half * 2;
        v2f aR  = *(const v2f*)(aC + kk);
        v2f bRv = *(const v2f*)(bR + kk);
        v2f aI  = *(const v2f*)(aS + kk);
        v2f bIv = *(const v2f*)(bI + kk);
        cr = __builtin_amdgcn_wmma_f32_16x16x4_f32(
                 false, aR, false, bRv, (short)0, cr, false, false);
        ci = __builtin_amdgcn_wmma_f32_16x16x4_f32(
                 false, aI, false, bIv, (short)0, ci, false, false);
    }

#pragma unroll
    for (int r = 0; r < 8; ++r) {
        const int row = mTile * 16 + r + half * 8;
        outReal[row * OUT_F + o] = cr[r];
        outImag[row * OUT_F + o] = ci[r];
    }
}

// ---------------------------------------------------------------------------
extern "C" void kernel_launch(void* const* d_in, const int* in_sizes, int n_in,
                              void* d_out, int out_size, void* d_ws, size_t ws_size,
                              hipStream_t stream) {
    const float* x  = (const float*)d_in[0];   // (256, 512)
    const float* W  = (const float*)d_in[1];   // (1024, 512)
    const float* B  = (const float*)d_in[2];   // (1024, 512)
    const float* Wr = (const float*)d_in[3];   // (512, 1024)
    const float* Wi = (const float*)d_in[4];   // (512, 1024)

    float* out  = (float*)d_out;               // [real | imag], each 256*512
    float* lut  = (float*)d_ws;                // 8192 f32 (sin | cos)
    float* Scos = lut + 2 * LUT;               // 256*1024 f32
    float* Ssin = Scos + BATCH * NNEUR;        // 256*1024 f32 (~2.06 MB total)

    build_lut_kernel<<<LUT / 256, 256, 0, stream>>>(lut);

    dim3 g1(NNEUR / 64, BATCH);                // 16 x 256 blocks, 8 waves each
    resonant_sums_kernel<<<g1, 256, 0, stream>>>(x, W, B, lut, Scos, Ssin);

    dim3 g2(OUT_F / 64, BATCH / 16);           // 8 x 16 blocks, 4 waves = 4 N-tiles
    resonant_gemm_kernel<<<g2, 128, 0, stream>>>(Scos, Ssin, Wr, Wi,
                                                 out, out + BATCH * OUT_F);
}